// DeepFM_40759239639138
// MI455X (gfx1250) — compile-verified
//
#include <hip/hip_runtime.h>
#include <stdint.h>

typedef __attribute__((ext_vector_type(2))) float v2f;
typedef __attribute__((ext_vector_type(8))) float v8f;

#define NF       39
#define DIMK     16
#define EOUT     624          // 39*16
#define ESTRIDE  644          // padded row stride (floats) for e_lds
#define XSTRIDE  40
#define HSTRIDE  17

__device__ __forceinline__ uint32_t lds_addr32(const void* p) {
    // generic pointers to LDS carry the AS(3) byte offset in addr[31:0]
    return (uint32_t)(uintptr_t)p;
}

__global__ __launch_bounds__(32)
void deepfm_tile16_kernel(const int* __restrict__ x,
                          const float* __restrict__ emb,
                          const float* __restrict__ linw,
                          const float* __restrict__ linb,
                          const float* __restrict__ W1, const float* __restrict__ b1,
                          const float* __restrict__ g1, const float* __restrict__ be1,
                          const float* __restrict__ W2, const float* __restrict__ b2,
                          const float* __restrict__ g2, const float* __restrict__ be2,
                          const float* __restrict__ W3, const float* __restrict__ b3,
                          float* __restrict__ out)
{
    __shared__ __align__(16) float e_lds[16 * ESTRIDE]; // 16 rows x 624 (padded)
    __shared__ float h_lds[16 * HSTRIDE];               // 16x16 hidden bounce (padded)
    __shared__ float lin_part[16 * XSTRIDE];
    __shared__ float fm_part[256];
    __shared__ int   x_lds[16 * XSTRIDE];

    const int lane = threadIdx.x;           // 0..31 (wave32)
    const int row0 = blockIdx.x * 16;

    // ---- stage index tile ----
    for (int t = lane; t < 16 * NF; t += 32) {
        int r = t / NF, f = t - r * NF;
        x_lds[r * XSTRIDE + f] = x[(row0 + r) * NF + f];
    }
    __syncthreads();

    // ---- async gather: memory -> LDS directly (ASYNCcnt path) ----
    // one (row,field) pair per lane-iteration: 4x B128 covers the 64B embedding
    // row (INST_OFFSET applies to BOTH global and LDS address), 1x B32 for lin_w
    for (int pair = lane; pair < 16 * NF; pair += 32) {
        int r = pair / NF, f = pair - r * NF;
        int g = x_lds[r * XSTRIDE + f] + f * 50000;     // OFFSETS[f] = f*50000
        const float* ga = emb + (size_t)g * DIMK;
        uint32_t     la = lds_addr32(&e_lds[r * ESTRIDE + f * DIMK]);
        asm volatile(
            "global_load_async_to_lds_b128 %0, %1, off\n\t"
            "global_load_async_to_lds_b128 %0, %1, off offset:16\n\t"
            "global_load_async_to_lds_b128 %0, %1, off offset:32\n\t"
            "global_load_async_to_lds_b128 %0, %1, off offset:48"
            :: "v"(la), "v"(ga) : "memory");
        const float* gw = linw + g;
        uint32_t     ll = lds_addr32(&lin_part[r * XSTRIDE + f]);
        asm volatile(
            "global_load_async_to_lds_b32 %0, %1, off"
            :: "v"(ll), "v"(gw) : "memory");
    }
#if __has_builtin(__builtin_amdgcn_s_wait_asynccnt)
    __builtin_amdgcn_s_wait_asynccnt(0);
#else
    asm volatile("s_wait_asynccnt 0x0" ::: "memory");
#endif
    __syncthreads();

    // ---- FM second-order partials: per (row, d): s^2 - sum_f e^2 ----
    for (int p = lane; p < 256; p += 32) {
        int r = p >> 4, d = p & 15;
        float s = 0.f, ssq = 0.f;
        const float* er = &e_lds[r * ESTRIDE + d];
        #pragma unroll
        for (int f = 0; f < NF; ++f) { float v = er[f * DIMK]; s += v; ssq += v * v; }
        fm_part[p] = s * s - ssq;
    }
    __syncthreads();

    // ---- MLP layer 1: (16 x 624) @ (624 x 16) via 156 x V_WMMA_F32_16X16X4_F32 ----
    const int half = lane >> 4;      // which half-wave
    const int ml   = lane & 15;      // A row m / B,C column n
    const float rs = rsqrtf(1.0f + 1e-5f);          // BN eval: mean 0, var 1
    const float a1 = g1[ml] * rs, s1 = b1[ml] * a1 + be1[ml];
    const float a2 = g2[ml] * rs, s2 = b2[ml] * a2 + be2[ml];

    v8f acc = {};
    const float* erow = &e_lds[ml * ESTRIDE];
    for (int kb = 0; kb < EOUT / 4; ++kb) {
        int k0 = kb * 4 + half * 2;
        v2f a; a.x = erow[k0];            a.y = erow[k0 + 1];              // A[m][k0..k0+1]
        v2f b; b.x = W1[k0 * DIMK + ml];  b.y = W1[(k0 + 1) * DIMK + ml];  // B[k][n]
        acc = __builtin_amdgcn_wmma_f32_16x16x4_f32(
            false, a, false, b, (short)0, acc, false, false);
    }
    // fold BN + ReLU (channel = ml), write h1 to LDS (C layout: vgpr r -> m = r + 8*half)
    #pragma unroll
    for (int r = 0; r < 8; ++r) {
        float h = acc[r] * a1 + s1;
        h_lds[(r + half * 8) * HSTRIDE + ml] = h > 0.f ? h : 0.f;
    }
    __syncthreads();

    // ---- MLP layer 2: (16x16) @ (16x16), 4 WMMAs ----
    v8f acc2 = {};
    #pragma unroll
    for (int kb = 0; kb < 4; ++kb) {
        int k0 = kb * 4 + half * 2;
        v2f a; a.x = h_lds[ml * HSTRIDE + k0]; a.y = h_lds[ml * HSTRIDE + k0 + 1];
        v2f b; b.x = W2[k0 * DIMK + ml];       b.y = W2[(k0 + 1) * DIMK + ml];
        acc2 = __builtin_amdgcn_wmma_f32_16x16x4_f32(
            false, a, false, b, (short)0, acc2, false, false);
    }
    __syncthreads();
    #pragma unroll
    for (int r = 0; r < 8; ++r) {
        float h = acc2[r] * a2 + s2;
        h_lds[(r + half * 8) * HSTRIDE + ml] = h > 0.f ? h : 0.f;
    }
    __syncthreads();

    // ---- layer 3 (16 -> 1) + lin + fm, lanes 0..15 = rows ----
    if (lane < 16) {
        int r = lane;
        float lin = linb[0];
        #pragma unroll
        for (int f = 0; f < NF; ++f) lin += lin_part[r * XSTRIDE + f];
        float fm = 0.f;
        #pragma unroll
        for (int d = 0; d < 16; ++d) fm += fm_part[r * 16 + d];
        fm *= 0.5f;
        float mlp = b3[0];
        #pragma unroll
        for (int k = 0; k < 16; ++k) mlp += h_lds[r * HSTRIDE + k] * W3[k];
        out[row0 + r] = lin + fm + mlp;
    }
}

extern "C" void kernel_launch(void* const* d_in, const int* in_sizes, int n_in,
                              void* d_out, int out_size, void* d_ws, size_t ws_size,
                              hipStream_t stream) {
    const int*   x    = (const int*)  d_in[0];
    const float* emb  = (const float*)d_in[1];
    const float* linw = (const float*)d_in[2];
    const float* linb = (const float*)d_in[3];
    const float* W1   = (const float*)d_in[4];
    const float* b1   = (const float*)d_in[5];
    const float* g1   = (const float*)d_in[6];
    const float* be1  = (const float*)d_in[7];
    const float* W2   = (const float*)d_in[8];
    const float* b2   = (const float*)d_in[9];
    const float* g2   = (const float*)d_in[10];
    const float* be2  = (const float*)d_in[11];
    const float* W3   = (const float*)d_in[12];
    const float* b3   = (const float*)d_in[13];
    float* out = (float*)d_out;

    const int batch = in_sizes[0] / NF;      // 16384
    const int tiles = batch / 16;            // 1024
    deepfm_tile16_kernel<<<tiles, 32, 0, stream>>>(
        x, emb, linw, linb, W1, b1, g1, be1, W2, b2, g2, be2, W3, b3, out);
}